// MultiChoiceCE_12128987644159
// MI455X (gfx1250) — compile-verified
//
#include <hip/hip_runtime.h>
#include <hip/hip_bf16.h>
#include <stdint.h>

#define NIMG 8
#define NCLS 19
#define HW   (512*512)
#define NSP  2048
#define ROWF 20                               // target row stride (last col sliced off)
#define TBL_BYTES (NSP*ROWF*4)                // 163840 B: whole per-image target table
#define PIX_PER_BLOCK 8192
#define CHUNKS_PER_IMG (HW / PIX_PER_BLOCK)   // 32
#define NBLOCKS (NIMG * CHUNKS_PER_IMG)       // 256
#define THREADS 256
#define ITERS (PIX_PER_BLOCK / (THREADS*4))   // 8 (4 pixels per thread per iter)

typedef float v2f __attribute__((ext_vector_type(2)));
typedef float v8f __attribute__((ext_vector_type(8)));

__global__ __launch_bounds__(THREADS)
void mcce_main(const float* __restrict__ inputs,
               const float* __restrict__ targets,
               const int*   __restrict__ superpixels,
               const unsigned char* __restrict__ spmasks,
               float* __restrict__ partials)
{
  extern __shared__ char smem[];
  float* tbl = (float*)smem;                  // 2048 x 20 f32 target table
  float* red = (float*)(smem + TBL_BYTES);    // per-wave reduction scratch

  const int tid   = threadIdx.x;
  const int n     = blockIdx.x / CHUNKS_PER_IMG;
  const int chunk = blockIdx.x % CHUNKS_PER_IMG;
  const int p0    = chunk * PIX_PER_BLOCK;

  // ---- stage image-n target table into LDS via gfx1250 async DMA loads ----
  {
    uint64_t g = (uint64_t)(uintptr_t)(targets + (size_t)n * NSP * ROWF)
               + (uint64_t)tid * 16u;
    uint32_t l = (uint32_t)(uintptr_t)tbl + (uint32_t)tid * 16u;
#pragma unroll
    for (int i = 0; i < TBL_BYTES / (THREADS * 16); ++i) {  // 40 x 4KB
      asm volatile("global_load_async_to_lds_b128 %0, %1, off"
                   :: "v"(l), "v"(g) : "memory");
      g += THREADS * 16;
      l += THREADS * 16;
    }
    asm volatile("s_wait_asynccnt 0" ::: "memory");
  }
  __syncthreads();

  const float4* tbl4 = (const float4*)tbl;    // rows are 80B = 16B aligned
  const float* inBase = inputs + (size_t)n * NCLS * HW;

  float lossAcc = 0.f;
  float cntAcc  = 0.f;

  for (int it = 0; it < ITERS; ++it) {
    const int p = p0 + (it * THREADS + tid) * 4;   // 4 consecutive pixels

    float4 r[NCLS];
#pragma unroll
    for (int c = 0; c < NCLS; ++c)
      r[c] = *(const float4*)(inBase + (size_t)c * HW + p);   // 512B/wave, coalesced

    int4     sidx  = *(const int4*)(superpixels + (size_t)n * HW + p);
    uint32_t mbits = *(const uint32_t*)(spmasks + (size_t)n * HW + p);

#pragma unroll
    for (int j = 0; j < 4; ++j) {
      float x[NCLS];
      float m = -3.4e38f;
#pragma unroll
      for (int c = 0; c < NCLS; ++c) {
        float4 v = r[c];
        float xv = (j == 0) ? v.x : (j == 1) ? v.y : (j == 2) ? v.z : v.w;
        x[c] = xv;
        m = fmaxf(m, xv);
      }
      int s = (j == 0) ? sidx.x : (j == 1) ? sidx.y : (j == 2) ? sidx.z : sidx.w;
      const float4* trow = tbl4 + (size_t)s * (ROWF / 4);     // 5 x ds_load_b128
      float4 t0 = trow[0], t1 = trow[1], t2 = trow[2], t3 = trow[3], t4v = trow[4];
      float t[NCLS] = { t0.x,t0.y,t0.z,t0.w, t1.x,t1.y,t1.z,t1.w,
                        t2.x,t2.y,t2.z,t2.w, t3.x,t3.y,t3.z,t3.w,
                        t4v.x,t4v.y,t4v.z };
      float denom = 0.f, numer = 0.f, tsum = 0.f;
#pragma unroll
      for (int c = 0; c < NCLS; ++c) {
        float e = __expf(x[c] - m);
        denom += e;
        numer += e * t[c];
        tsum  += t[c];                        // targets are {0,1}: any!=0 <=> sum>0
      }
      bool valid = (((mbits >> (8 * j)) & 0xffu) != 0u) && (tsum > 0.f);
      // log(numer/denom + eps) == log((numer + eps*denom)/denom)
      float term = __logf(numer + 1e-8f * denom) - __logf(denom);
      lossAcc += valid ? term : 0.f;
      cntAcc  += valid ? 1.f  : 0.f;
    }
  }

  // ---- wave32 (loss,cnt) reduction via V_WMMA_F32_16X16X4_F32 ----
  // A 16x4 layout: lane L supplies {vgpr0,vgpr1} = rows M=L%16, K={0,1} (L<16) or K={2,3} (L>=16)
  //   => A[i][:] = { loss_i, cnt_i, loss_{i+16}, cnt_{i+16} }
  // B 4x16: column 0 = (1,0,1,0), column 1 = (0,1,0,1)
  //   => D[i][0] = loss_i + loss_{i+16},  D[i][1] = cnt_i + cnt_{i+16}
  const unsigned lane = threadIdx.x & 31u;
  const unsigned l16  = lane & 15u;
  v2f a; a.x = lossAcc; a.y = cntAcc;
  v2f b; b.x = (l16 == 0u) ? 1.f : 0.f;       // B vgpr0: K=0 | K=2
        b.y = (l16 == 1u) ? 1.f : 0.f;        // B vgpr1: K=1 | K=3
  v8f cacc = {};
  cacc = __builtin_amdgcn_wmma_f32_16x16x4_f32(
      /*neg_a=*/false, a, /*neg_b=*/false, b,
      /*c_mod=*/(short)0, cacc, /*reuse_a=*/false, /*reuse_b=*/false);
  float srow = cacc[0]+cacc[1]+cacc[2]+cacc[3]+cacc[4]+cacc[5]+cacc[6]+cacc[7];
  // column 0 (loss) lives in lanes 0 (M=0..7) and 16 (M=8..15); column 1 (cnt) in lanes 1,17
  float lossW = __shfl(srow, 0, 32) + __shfl(srow, 16, 32);
  float cntW  = __shfl(srow, 1, 32) + __shfl(srow, 17, 32);

  const int wave = threadIdx.x >> 5;
  if (lane == 0) { red[2*wave] = lossW; red[2*wave+1] = cntW; }
  __syncthreads();
  if (tid == 0) {
    float L = 0.f, Cn = 0.f;
    for (int w = 0; w < THREADS/32; ++w) { L += red[2*w]; Cn += red[2*w+1]; }
    partials[2*blockIdx.x]   = L;
    partials[2*blockIdx.x+1] = Cn;
  }
}

__global__ __launch_bounds__(256)
void mcce_finalize(const float* __restrict__ partials, float* __restrict__ out)
{
  __shared__ float sl[256], sc[256];
  int t = threadIdx.x;
  sl[t] = partials[2*t];
  sc[t] = partials[2*t+1];
  __syncthreads();
  for (int s = 128; s > 0; s >>= 1) {
    if (t < s) { sl[t] += sl[t+s]; sc[t] += sc[t+s]; }
    __syncthreads();
  }
  if (t == 0) out[0] = -sl[0] / (1.f + sc[0]);   // loss / (1 + num_valid)
}

extern "C" void kernel_launch(void* const* d_in, const int* in_sizes, int n_in,
                              void* d_out, int out_size, void* d_ws, size_t ws_size,
                              hipStream_t stream) {
  const float*         inputs      = (const float*)d_in[0];
  const float*         targets     = (const float*)d_in[1];
  const int*           superpixels = (const int*)d_in[2];
  const unsigned char* spmasks     = (const unsigned char*)d_in[3]; // jax bool = 1 byte
  float* out = (float*)d_out;
  float* ws  = (float*)d_ws;       // 256 blocks x {loss, cnt} partials = 2 KB

  size_t shmem = TBL_BYTES + 64;   // target table + per-wave reduction scratch
  mcce_main<<<dim3(NBLOCKS), dim3(THREADS), shmem, stream>>>(
      inputs, targets, superpixels, spmasks, ws);
  mcce_finalize<<<dim3(1), dim3(256), 0, stream>>>(ws, out);
}